// StitchEncoder_81389630259656
// MI455X (gfx1250) — compile-verified
//
#include <hip/hip_runtime.h>

// ---------------------------------------------------------------------------
// StitchEncoder for MI455X (gfx1250, wave32, WMMA):
//   h   = softsign(x @ W1[eid] + b1[eid])        [B,F,2N]
//   out = h @ W2[eid] + b2[eid]                  [B,F,P]
// B=256 F=100 N=512 2N=1024 P=512 G=10, all fp32.
//
// Compute-bound (53.7 GFLOP vs ~150MB unique bytes; weights+x are L2
// resident). One workgroup per (trial, 16-row F tile); X tile and full h tile
// live in LDS; both GEMMs run on V_WMMA_F32_16X16X4_F32 (full fp32).
// Round-3 changes:
//  * 4-column-tile blocking per K-pass: one A ds_load feeds 4 WMMAs,
//    16 WMMAs per unrolled (K=16) iteration, 4 independent acc chains.
// ---------------------------------------------------------------------------

typedef __attribute__((ext_vector_type(2))) float v2f;
typedef __attribute__((ext_vector_type(4))) float v4f;
typedef __attribute__((ext_vector_type(8))) float v8f;

#define B_TRIALS 256
#define F_BINS   100
#define N_UNITS  512
#define M_HID    1024
#define P_OUT    512
#define FT       16          // F rows per tile
#define NFT      7           // ceil(100/16)
#define XS_STRIDE 516        // 512 + 4 pad -> conflict-free LDS, 16B aligned rows
#define HS_STRIDE 1028       // 1024 + 4 pad

__global__ __launch_bounds__(256) void stitch_encoder_kernel(
    const float* __restrict__ x,    // [B,F,N]
    const int*   __restrict__ eid,  // [B]
    const float* __restrict__ W1,   // [G,N,2N]
    const float* __restrict__ b1,   // [G,2N]
    const float* __restrict__ W2,   // [G,2N,P]
    const float* __restrict__ b2,   // [G,P]
    float*       __restrict__ out)  // [B,F,P]
{
    extern __shared__ float lds[];
    float* Xs = lds;                    // [FT][XS_STRIDE]
    float* Hs = lds + FT * XS_STRIDE;   // [FT][HS_STRIDE]

    const int blk  = blockIdx.x;
    const int b    = blk / NFT;
    const int ft   = blk % NFT;
    const int f0   = ft * FT;
    const int rows = (F_BINS - f0 < FT) ? (F_BINS - f0) : FT;

    const int g = eid[b];
    const float* W1g = W1 + (size_t)g * N_UNITS * M_HID;
    const float* b1g = b1 + (size_t)g * M_HID;
    const float* W2g = W2 + (size_t)g * M_HID * P_OUT;
    const float* b2g = b2 + (size_t)g * P_OUT;

    const int t = threadIdx.x;

    // ---- Stage X tile [16 x 512] into LDS (zero-pad rows past F) ----------
    const float* xb = x + ((size_t)b * F_BINS + f0) * N_UNITS;
    #pragma unroll
    for (int i = 0; i < 8; ++i) {
        int flat4 = t + i * 256;        // float4 index in [0, 2048)
        int row   = flat4 >> 7;         // 128 float4 per row
        int c4    = flat4 & 127;
        v4f val = {};
        if (row < rows)
            val = *(const v4f*)(xb + (size_t)row * N_UNITS + c4 * 4);
        *(v4f*)(Xs + row * XS_STRIDE + c4 * 4) = val;
    }
    __syncthreads();

    const int lane = t & 31;            // wave32
    const int wv   = t >> 5;            // 0..7
    const int half = lane >> 4;         // 0/1  (selects K pair per ISA layout)
    const int m    = lane & 15;         // matrix row (A) / column (B,C,D)

    // ---- GEMM1: H = softsign(X @ W1g + b1g), H kept in LDS ----------------
    // wave wv owns N-tiles [wv*8, wv*8+8), processed as 2 groups of 4.
    for (int ntp = 0; ntp < 2; ++ntp) {
        const int col0 = (wv * 8 + ntp * 4) * 16;
        v8f acc0 = {}, acc1 = {}, acc2 = {}, acc3 = {};
        const float* Arow = Xs + m * XS_STRIDE + 2 * half;
        const float* Bcol = W1g + (size_t)(2 * half) * M_HID + col0 + m;
        #pragma unroll 4
        for (int k0 = 0; k0 < N_UNITS; k0 += 4) {
            // A: 16x4 tile; lane holds (K = k0+2*half, +1) of row m.
            // Shared by all 4 column tiles.
            v2f a = *(const v2f*)(Arow + k0);
            // B: 4x16 tiles; lane holds rows (k0+2*half, +1) of col+m
            const float* Bk0 = Bcol + (size_t)k0 * M_HID;
            const float* Bk1 = Bcol + (size_t)(k0 + 1) * M_HID;
            v2f bb0, bb1, bb2, bb3;
            bb0.x = Bk0[ 0]; bb0.y = Bk1[ 0];
            bb1.x = Bk0[16]; bb1.y = Bk1[16];
            bb2.x = Bk0[32]; bb2.y = Bk1[32];
            bb3.x = Bk0[48]; bb3.y = Bk1[48];
            acc0 = __builtin_amdgcn_wmma_f32_16x16x4_f32(
                false, a, false, bb0, (short)0, acc0, false, false);
            acc1 = __builtin_amdgcn_wmma_f32_16x16x4_f32(
                false, a, false, bb1, (short)0, acc1, false, false);
            acc2 = __builtin_amdgcn_wmma_f32_16x16x4_f32(
                false, a, false, bb2, (short)0, acc2, false, false);
            acc3 = __builtin_amdgcn_wmma_f32_16x16x4_f32(
                false, a, false, bb3, (short)0, acc3, false, false);
        }
        const float bias0 = b1g[col0 + m];
        const float bias1 = b1g[col0 + 16 + m];
        const float bias2 = b1g[col0 + 32 + m];
        const float bias3 = b1g[col0 + 48 + m];
        float* Hrow = Hs + (8 * half) * HS_STRIDE + col0 + m;
        #pragma unroll
        for (int r = 0; r < 8; ++r) {
            // softsign: denominator 1+|h| >= 1, so fast rcp is safe (~1 ulp)
            float h0 = acc0[r] + bias0;
            float h1 = acc1[r] + bias1;
            float h2 = acc2[r] + bias2;
            float h3 = acc3[r] + bias3;
            h0 = h0 * __builtin_amdgcn_rcpf(1.0f + __builtin_fabsf(h0));
            h1 = h1 * __builtin_amdgcn_rcpf(1.0f + __builtin_fabsf(h1));
            h2 = h2 * __builtin_amdgcn_rcpf(1.0f + __builtin_fabsf(h2));
            h3 = h3 * __builtin_amdgcn_rcpf(1.0f + __builtin_fabsf(h3));
            Hrow[r * HS_STRIDE +  0] = h0;
            Hrow[r * HS_STRIDE + 16] = h1;
            Hrow[r * HS_STRIDE + 32] = h2;
            Hrow[r * HS_STRIDE + 48] = h3;
        }
    }
    __syncthreads();

    // ---- GEMM2: out = H @ W2g + b2g ---------------------------------------
    // wave wv owns P-tiles [wv*4, wv*4+4), one group of 4.
    float* outb = out + ((size_t)b * F_BINS + f0) * P_OUT;
    {
        const int col0 = (wv * 4) * 16;
        v8f acc0 = {}, acc1 = {}, acc2 = {}, acc3 = {};
        const float* Arow = Hs + m * HS_STRIDE + 2 * half;
        const float* Bcol = W2g + (size_t)(2 * half) * P_OUT + col0 + m;
        #pragma unroll 4
        for (int k0 = 0; k0 < M_HID; k0 += 4) {
            v2f a = *(const v2f*)(Arow + k0);
            const float* Bk0 = Bcol + (size_t)k0 * P_OUT;
            const float* Bk1 = Bcol + (size_t)(k0 + 1) * P_OUT;
            v2f bb0, bb1, bb2, bb3;
            bb0.x = Bk0[ 0]; bb0.y = Bk1[ 0];
            bb1.x = Bk0[16]; bb1.y = Bk1[16];
            bb2.x = Bk0[32]; bb2.y = Bk1[32];
            bb3.x = Bk0[48]; bb3.y = Bk1[48];
            acc0 = __builtin_amdgcn_wmma_f32_16x16x4_f32(
                false, a, false, bb0, (short)0, acc0, false, false);
            acc1 = __builtin_amdgcn_wmma_f32_16x16x4_f32(
                false, a, false, bb1, (short)0, acc1, false, false);
            acc2 = __builtin_amdgcn_wmma_f32_16x16x4_f32(
                false, a, false, bb2, (short)0, acc2, false, false);
            acc3 = __builtin_amdgcn_wmma_f32_16x16x4_f32(
                false, a, false, bb3, (short)0, acc3, false, false);
        }
        const float bias0 = b2g[col0 + m];
        const float bias1 = b2g[col0 + 16 + m];
        const float bias2 = b2g[col0 + 32 + m];
        const float bias3 = b2g[col0 + 48 + m];
        #pragma unroll
        for (int r = 0; r < 8; ++r) {
            int fr = r + 8 * half;
            if (fr < rows) {
                float* orow = outb + (size_t)fr * P_OUT + col0 + m;
                orow[ 0] = acc0[r] + bias0;
                orow[16] = acc1[r] + bias1;
                orow[32] = acc2[r] + bias2;
                orow[48] = acc3[r] + bias3;
            }
        }
    }
}

extern "C" void kernel_launch(void* const* d_in, const int* in_sizes, int n_in,
                              void* d_out, int out_size, void* d_ws, size_t ws_size,
                              hipStream_t stream) {
    (void)in_sizes; (void)n_in; (void)out_size; (void)d_ws; (void)ws_size;
    const float* x   = (const float*)d_in[0];
    const int*   eid = (const int*)  d_in[1];
    const float* W1  = (const float*)d_in[2];
    const float* b1  = (const float*)d_in[3];
    const float* W2  = (const float*)d_in[4];
    const float* b2  = (const float*)d_in[5];
    float* out = (float*)d_out;

    const size_t shmem = (size_t)(FT * XS_STRIDE + FT * HS_STRIDE) * sizeof(float); // ~99 KB
    dim3 grid(B_TRIALS * NFT);   // 1792 workgroups
    dim3 block(256);             // 8 wave32 per workgroup
    hipLaunchKernelGGL(stitch_encoder_kernel, grid, block, shmem, stream,
                       x, eid, W1, b1, W2, b2, out);
}